// Ive_3736621547854
// MI455X (gfx1250) — compile-verified
//
#include <hip/hip_runtime.h>
#include <hip/hip_bf16.h>

// i1e(z) = I_1(z) * exp(-|z|)  (exponentially scaled modified Bessel, v=1)
// 8192x8192 fp32 elementwise map -> strictly HBM-bandwidth bound on MI455X
// (512 MB traffic @ 23.3 TB/s ~ 22 us floor). Strategy: float4 non-temporal
// streaming loads/stores, branchless A&S 9.8.3/9.8.4 polynomial evaluation
// using hardware TRANS ops (v_exp_f32 / v_rcp_f32 / v_rsq_f32).

typedef float v4f __attribute__((ext_vector_type(4)));

__device__ __forceinline__ float i1e_eval(float z) {
    const float az = __builtin_fabsf(z);

    // ---- small branch: z in [0, 3.75):  I1(z)/z = P6(t), t=(z/3.75)^2 ----
    float ts = az * 0.26666666666666666f;   // z / 3.75
    ts = ts * ts;
    float ps = __builtin_fmaf(ts, 0.00032411f, 0.00301532f);
    ps = __builtin_fmaf(ts, ps, 0.02658733f);
    ps = __builtin_fmaf(ts, ps, 0.15084934f);
    ps = __builtin_fmaf(ts, ps, 0.51498869f);
    ps = __builtin_fmaf(ts, ps, 0.87890594f);
    ps = __builtin_fmaf(ts, ps, 0.5f);
    // exp(-az) = exp2(-az * log2(e))  -> v_exp_f32 on the TRANS pipe
    float e = __builtin_amdgcn_exp2f(az * -1.4426950408889634f);
    float small_r = az * ps * e;

    // ---- large branch: z >= 3.75:  sqrt(z)*e^-z*I1(z) = Q8(t), t=3.75/z ----
    float tl = 3.75f * __builtin_amdgcn_rcpf(az);
    float pl = __builtin_fmaf(tl, -0.00420059f, 0.01787654f);
    pl = __builtin_fmaf(tl, pl, -0.02895312f);
    pl = __builtin_fmaf(tl, pl, 0.02282967f);
    pl = __builtin_fmaf(tl, pl, -0.01031555f);
    pl = __builtin_fmaf(tl, pl, 0.00163801f);
    pl = __builtin_fmaf(tl, pl, -0.00362018f);
    pl = __builtin_fmaf(tl, pl, -0.03988024f);
    pl = __builtin_fmaf(tl, pl, 0.39894228f);
    float large_r = pl * __builtin_amdgcn_rsqf(az);

    float r = (az < 3.75f) ? small_r : large_r;
    // i1e is odd in z (inputs here are positive, but keep it correct)
    return __builtin_copysignf(r, z);
}

__global__ __launch_bounds__(256) void Ive_i1e_kernel(const float* __restrict__ z,
                                                      float* __restrict__ out,
                                                      long long n) {
    long long i = ((long long)blockIdx.x * blockDim.x + threadIdx.x) * 4LL;
    if (i + 3 < n) {
        // 128-bit streaming load with non-temporal hint (working set > L2)
        v4f v = __builtin_nontemporal_load((const v4f*)(z + i));
        v4f r;
        r.x = i1e_eval(v.x);
        r.y = i1e_eval(v.y);
        r.z = i1e_eval(v.z);
        r.w = i1e_eval(v.w);
        __builtin_nontemporal_store(r, (v4f*)(out + i));
    } else if (i < n) {
        // scalar tail (n not multiple of 4)
        for (; i < n; ++i) {
            out[i] = i1e_eval(z[i]);
        }
    }
}

extern "C" void kernel_launch(void* const* d_in, const int* in_sizes, int n_in,
                              void* d_out, int out_size, void* d_ws, size_t ws_size,
                              hipStream_t stream) {
    (void)n_in; (void)d_ws; (void)ws_size; (void)out_size;
    const float* z = (const float*)d_in[0];
    float* out = (float*)d_out;
    long long n = (long long)in_sizes[0];

    long long nvec = (n + 3) / 4;                 // one float4 per thread
    int block = 256;                               // 8 wave32 waves per block
    long long grid = (nvec + block - 1) / block;   // ~65536 blocks for 64M elems

    Ive_i1e_kernel<<<(dim3)(unsigned)grid, (dim3)block, 0, stream>>>(z, out, n);
}